// GoodwinVolterra_24464133718448
// MI455X (gfx1250) — compile-verified
//
#include <hip/hip_runtime.h>

// Goodwin-Lotka-Volterra Euler integration.
//
// Sequential 2-state nonlinear ODE: no matrix structure, so WMMA/TDM are
// inapplicable. Total work ~4 KFLOP, output 1.6 KB -> pure launch-latency
// bound on MI455X. Strategy: single 128-thread (4 x wave32) block; lane 0 runs the dependent
// chain in VGPRs, stages clamped trajectory values in LDS, then all lanes
// flush LDS -> global with coalesced stores.

#define GV_ALPHA 0.1f
#define GV_BETA  0.5f
#define GV_DELTA 0.5f
#define GV_GAMMA 0.2f
#define GV_DT    0.1f
#define GV_X0    0.8f
#define GV_Y0    0.3f

// LDS staging capacity (per trajectory). 2 * 2048 * 4B = 16 KB LDS,
// far under the 320 KB/WGP budget. steps beyond this fall back to
// direct global stores from lane 0 (never hit for steps == 200).
#define GV_MAX_LDS_STEPS 2048

__global__ void GoodwinVolterra_kernel(const float* __restrict__ ps_ptr,
                                       const int* __restrict__ steps_ptr,
                                       float* __restrict__ out) {
    __shared__ float s_traj[2 * GV_MAX_LDS_STEPS];

    const int   steps = steps_ptr[0];
    const float ps    = fabsf(ps_ptr[0]);

    if (threadIdx.x == 0) {
        float x = GV_X0;
        float y = GV_Y0;
        for (int i = 0; i < steps; ++i) {
            const float xy           = x * y;
            const float intervention = -ps * xy;
            const float dx = GV_ALPHA * x - GV_BETA * xy + intervention;
            const float dy = GV_DELTA * xy - GV_GAMMA * y - intervention;
            // Unclamped state propagates (matches the reference exactly).
            x = x + dx * GV_DT;
            y = y + dy * GV_DT;
            // Clamped values are what gets recorded.
            const float xc = fminf(fmaxf(x, 0.0f), 1.0f);
            const float yc = fminf(fmaxf(y, 0.0f), 1.0f);
            if (i < GV_MAX_LDS_STEPS) {
                s_traj[i]                    = xc;
                s_traj[GV_MAX_LDS_STEPS + i] = yc;
            } else {
                // Overflow fallback: direct global store (not hit at steps=200).
                out[i]         = xc;
                out[steps + i] = yc;
            }
        }
    }

    __syncthreads();

    // Coalesced cooperative flush: xs then ys, concatenated flat.
    const int n = steps < GV_MAX_LDS_STEPS ? steps : GV_MAX_LDS_STEPS;
    for (int i = (int)threadIdx.x; i < n; i += (int)blockDim.x) {
        out[i]         = s_traj[i];
        out[steps + i] = s_traj[GV_MAX_LDS_STEPS + i];
    }
}

extern "C" void kernel_launch(void* const* d_in, const int* in_sizes, int n_in,
                              void* d_out, int out_size, void* d_ws, size_t ws_size,
                              hipStream_t stream) {
    (void)in_sizes; (void)n_in; (void)d_ws; (void)ws_size; (void)out_size;

    const float* ps    = (const float*)d_in[0];
    const int*   steps = (const int*)d_in[1];
    float*       out   = (float*)d_out;

    // One workgroup: 4 wave32 waves. Wave 0 lane 0 runs the recurrence;
    // all 128 lanes participate in the coalesced LDS->global flush.
    GoodwinVolterra_kernel<<<dim3(1), dim3(128), 0, stream>>>(ps, steps, out);
}